// GCN_73323681677457
// MI455X (gfx1250) — compile-verified
//
#include <hip/hip_runtime.h>

typedef __attribute__((ext_vector_type(2))) float v2f;
typedef __attribute__((ext_vector_type(8))) float v8f;
typedef __attribute__((ext_vector_type(4))) unsigned int tdm_v4u;
typedef __attribute__((ext_vector_type(4))) int          tdm_v4i;
typedef __attribute__((ext_vector_type(8))) int          tdm_v8i;

// ---------------------------------------------------------------------------
// degree: deg[d] += 1 for every edge target (self-loop +1 added in dinv pass)
// ---------------------------------------------------------------------------
__global__ void gcn_deg_kernel(const int* __restrict__ dst, float* __restrict__ deg, int E) {
    int e = blockIdx.x * blockDim.x + threadIdx.x;
    if (e < E) atomicAdd(&deg[dst[e]], 1.0f);
}

// dinv[i] = rsqrt(deg[i] + 1)   (in place; deg >= 1 always due to self loop)
__global__ void gcn_dinv_kernel(float* __restrict__ deg, int N) {
    int i = blockIdx.x * blockDim.x + threadIdx.x;
    if (i < N) {
        float d = deg[i] + 1.0f;
        deg[i] = __frsqrt_rn(d);
    }
}

// ---------------------------------------------------------------------------
// GEMM1: h[N x 64] = x[N x K] @ W[K x 64]  with fp32 WMMA 16x16x4.
// One wave per 32-row tile (two 16x16 A tiles, 8 accumulators) x full 64 cols.
// W staged into LDS once per block via the Tensor Data Mover (TDM).
// ---------------------------------------------------------------------------
__global__ void __launch_bounds__(256)
gcn_gemm1_wmma(const float* __restrict__ x, const float* __restrict__ W,
               float* __restrict__ h, int N, int K) {
    extern __shared__ float sW[];            // K * 64 floats (125 KB for K=500)
    const int HD = 64;

#if __has_builtin(__builtin_amdgcn_tensor_load_to_lds)
    if (threadIdx.x < 32) {                  // one wave issues the TDM copy
        unsigned long long ga = (unsigned long long)(uintptr_t)W;
        unsigned T = (unsigned)(K * HD);     // 32000 elements (fits tile_dim0)
        tdm_v4u g0;
        g0[0] = 1u;                                                 // count=1 user D#
        g0[1] = 0u;                                                 // lds_addr = dyn-LDS base
        g0[2] = (unsigned)(ga & 0xFFFFFFFFu);                       // global_addr[31:0]
        g0[3] = (unsigned)((ga >> 32) & 0x01FFFFFFu) | (2u << 30);  // addr[56:32] | type=2
        tdm_v8i g1;
        g1[0] = 0x00020000;                                         // data_size = 4 B
        g1[1] = (int)((T & 0xFFFFu) << 16);                         // tensor_dim0[15:0]
        g1[2] = (int)(((T >> 16) & 0xFFFFu) | (1u << 16));          // dim0[31:16] | tensor_dim1=1
        g1[3] = (int)((T & 0xFFFFu) << 16);                         // tile_dim0 = T
        g1[4] = 0;                                                  // tile_dim1/2 unused
        g1[5] = (int)T;                                             // tensor_dim0_stride
        g1[6] = 0;
        g1[7] = 0;
        tdm_v4i g2 = {0, 0, 0, 0};
        tdm_v4i g3 = {0, 0, 0, 0};
#if __clang_major__ >= 23
        tdm_v8i g4 = {0, 0, 0, 0, 0, 0, 0, 0};
        __builtin_amdgcn_tensor_load_to_lds(g0, g1, g2, g3, g4, 0);
#else
        __builtin_amdgcn_tensor_load_to_lds(g0, g1, g2, g3, 0);
#endif
        __builtin_amdgcn_s_wait_tensorcnt(0);
    }
#else
    for (int i = threadIdx.x; i < K * HD; i += blockDim.x) sW[i] = W[i];
#endif
    __syncthreads();

    int wave = (blockIdx.x * blockDim.x + threadIdx.x) >> 5;
    int lane = threadIdx.x & 31;
    int m0 = wave * 32;
    if (m0 >= N) return;                     // wave-uniform: EXEC stays all-1s

    int col = lane & 15;                     // tile column / A row
    int khi = (lane >> 4) << 1;              // 0 (lanes 0-15) or 2 (lanes 16-31)

    v8f c00 = {}, c01 = {}, c02 = {}, c03 = {};   // rows m0 .. m0+15
    v8f c10 = {}, c11 = {}, c12 = {}, c13 = {};   // rows m0+16 .. m0+31
    const float* xrow0 = x + (size_t)(m0 + col) * K;
    const float* xrow1 = xrow0 + (size_t)16 * K;

    for (int k0 = 0; k0 + 3 < K; k0 += 4) {
        // Two A 16x4 fragments: VGPR0 = K(khi), VGPR1 = K(khi+1) of row M=col
        v2f a0, a1;
        a0.x = xrow0[k0 + khi];
        a0.y = xrow0[k0 + khi + 1];
        a1.x = xrow1[k0 + khi];
        a1.y = xrow1[k0 + khi + 1];
        // B 4x16 fragments from LDS: rows K(khi)/K(khi+1), cols t*16 + col
        const float* wb = sW + (size_t)(k0 + khi) * HD;
        v2f b0, b1, b2, b3;
        b0.x = wb[ 0 + col]; b0.y = wb[HD +  0 + col];
        b1.x = wb[16 + col]; b1.y = wb[HD + 16 + col];
        b2.x = wb[32 + col]; b2.y = wb[HD + 32 + col];
        b3.x = wb[48 + col]; b3.y = wb[HD + 48 + col];
        c00 = __builtin_amdgcn_wmma_f32_16x16x4_f32(false, a0, false, b0, (short)0, c00, false, false);
        c01 = __builtin_amdgcn_wmma_f32_16x16x4_f32(false, a0, false, b1, (short)0, c01, false, false);
        c02 = __builtin_amdgcn_wmma_f32_16x16x4_f32(false, a0, false, b2, (short)0, c02, false, false);
        c03 = __builtin_amdgcn_wmma_f32_16x16x4_f32(false, a0, false, b3, (short)0, c03, false, false);
        c10 = __builtin_amdgcn_wmma_f32_16x16x4_f32(false, a1, false, b0, (short)0, c10, false, false);
        c11 = __builtin_amdgcn_wmma_f32_16x16x4_f32(false, a1, false, b1, (short)0, c11, false, false);
        c12 = __builtin_amdgcn_wmma_f32_16x16x4_f32(false, a1, false, b2, (short)0, c12, false, false);
        c13 = __builtin_amdgcn_wmma_f32_16x16x4_f32(false, a1, false, b3, (short)0, c13, false, false);
    }

    // C/D layout: VGPR i -> row (i or i+8), col = lane&15
    int mrow = m0 + ((lane >> 4) << 3);
    float* hr0 = h + (size_t)mrow * HD + col;
    float* hr1 = hr0 + (size_t)16 * HD;
#pragma unroll
    for (int i = 0; i < 8; i++) {
        hr0[(size_t)i * HD +  0] = c00[i];
        hr0[(size_t)i * HD + 16] = c01[i];
        hr0[(size_t)i * HD + 32] = c02[i];
        hr0[(size_t)i * HD + 48] = c03[i];
        hr1[(size_t)i * HD +  0] = c10[i];
        hr1[(size_t)i * HD + 16] = c11[i];
        hr1[(size_t)i * HD + 32] = c12[i];
        hr1[(size_t)i * HD + 48] = c13[i];
    }
}

// Fallback GEMM (only used for unexpected shapes): one thread per output element.
__global__ void gcn_gemm1_naive(const float* __restrict__ x, const float* __restrict__ W,
                                float* __restrict__ h, int N, int K, int H) {
    long long t = (long long)blockIdx.x * blockDim.x + threadIdx.x;
    if (t >= (long long)N * H) return;
    int n = (int)(t / H), j = (int)(t % H);
    float acc = 0.0f;
    for (int k = 0; k < K; k++) acc += x[(size_t)n * K + k] * W[(size_t)k * H + j];
    h[t] = acc;
}

// ---------------------------------------------------------------------------
// Propagation layer 1: wave32 per edge (incl. self loops), 2 features / lane.
// hagg[dst] += h[src] * dinv[src]*dinv[dst]
// ---------------------------------------------------------------------------
__global__ void gcn_scatter64_kernel(const int* __restrict__ src, const int* __restrict__ dst,
                                     const float* __restrict__ dinv, const float* __restrict__ h,
                                     float* __restrict__ hagg, int E, int N) {
    long long t = (long long)blockIdx.x * blockDim.x + threadIdx.x;
    long long edge = t >> 5;
    int lane = (int)(t & 31);
    if (edge >= (long long)E + N) return;
    int s, d;
    if (edge < E) { s = src[edge]; d = dst[edge]; }
    else          { s = d = (int)(edge - E); }                 // self loop
    float nrm = dinv[s] * dinv[d];
    const float2 v = ((const float2*)(h + (size_t)s * 64))[lane];
    float* hd = hagg + (size_t)d * 64 + 2 * lane;
    atomicAdd(hd + 0, v.x * nrm);
    atomicAdd(hd + 1, v.y * nrm);
}

// ---------------------------------------------------------------------------
// Layer 2 transform fused with layer-1 bias + ReLU:
// g[n,c] = sum_k relu(hagg[n,k] + b1[k]) * W2[k,c]
// ---------------------------------------------------------------------------
__global__ void gcn_gemm2_kernel(const float* __restrict__ hagg, const float* __restrict__ b1,
                                 const float* __restrict__ W2, float* __restrict__ g,
                                 int N, int H, int C) {
    int n = blockIdx.x * blockDim.x + threadIdx.x;
    if (n >= N) return;
    float acc[4] = {0.f, 0.f, 0.f, 0.f};
    const float* hr = hagg + (size_t)n * H;
    for (int k = 0; k < H; k++) {
        float v = hr[k] + b1[k];
        v = v > 0.0f ? v : 0.0f;
        for (int c = 0; c < C && c < 4; c++) acc[c] += v * W2[(size_t)k * C + c];
    }
    for (int c = 0; c < C && c < 4; c++) g[(size_t)n * C + c] = acc[c];
}

// Propagation layer 2: thread per edge, C (=3) atomics.
__global__ void gcn_scatterC_kernel(const int* __restrict__ src, const int* __restrict__ dst,
                                    const float* __restrict__ dinv, const float* __restrict__ g,
                                    float* __restrict__ out, int E, int N, int C) {
    long long e = (long long)blockIdx.x * blockDim.x + threadIdx.x;
    if (e >= (long long)E + N) return;
    int s, d;
    if (e < E) { s = src[e]; d = dst[e]; }
    else       { s = d = (int)(e - E); }
    float nrm = dinv[s] * dinv[d];
    for (int c = 0; c < C; c++)
        atomicAdd(&out[(size_t)d * C + c], g[(size_t)s * C + c] * nrm);
}

// out += b2 (broadcast over nodes)
__global__ void gcn_bias2_kernel(float* __restrict__ out, const float* __restrict__ b2,
                                 int N, int C) {
    long long t = (long long)blockIdx.x * blockDim.x + threadIdx.x;
    if (t < (long long)N * C) out[t] += b2[(int)(t % C)];
}

// ---------------------------------------------------------------------------
extern "C" void kernel_launch(void* const* d_in, const int* in_sizes, int n_in,
                              void* d_out, int out_size, void* d_ws, size_t ws_size,
                              hipStream_t stream) {
    const float* x  = (const float*)d_in[0];
    const int*   ei = (const int*)d_in[1];   // edge_index [2, E]
    const float* W1 = (const float*)d_in[2];
    const float* b1 = (const float*)d_in[3];
    const float* W2 = (const float*)d_in[4];
    const float* b2 = (const float*)d_in[5];

    const int H = in_sizes[3];               // 64
    const int C = in_sizes[5];               // 3
    const int F = in_sizes[2] / H;           // 500
    const int N = in_sizes[0] / F;           // 100000
    const int E = in_sizes[1] / 2;           // 1600000
    const int* src = ei;
    const int* dst = ei + E;

    float* ws   = (float*)d_ws;
    float* deg  = ws;                              // N      (becomes dinv)
    float* h    = deg  + (size_t)N;                // N * H
    float* hagg = h    + (size_t)N * H;            // N * H
    float* g    = hagg + (size_t)N * H;            // N * C
    float* out  = (float*)d_out;

    hipMemsetAsync(deg,  0, (size_t)N * sizeof(float), stream);
    hipMemsetAsync(hagg, 0, (size_t)N * H * sizeof(float), stream);
    hipMemsetAsync(out,  0, (size_t)N * C * sizeof(float), stream);

    const int TB = 256;

    gcn_deg_kernel<<<(E + TB - 1) / TB, TB, 0, stream>>>(dst, deg, E);
    gcn_dinv_kernel<<<(N + TB - 1) / TB, TB, 0, stream>>>(deg, N);

    if (H == 64 && (N % 32) == 0 && (F % 4) == 0 && (F * 64) < 65536) {
        size_t shmem = (size_t)F * 64 * sizeof(float);   // 125 KB for F=500
        hipFuncSetAttribute((const void*)gcn_gemm1_wmma,
                            hipFuncAttributeMaxDynamicSharedMemorySize, (int)shmem);
        int waves  = N / 32;                              // 3125
        int blocks = (waves + 7) / 8;                     // 8 waves / block
        gcn_gemm1_wmma<<<blocks, TB, shmem, stream>>>(x, W1, h, N, F);
    } else {
        long long tot = (long long)N * H;
        gcn_gemm1_naive<<<(int)((tot + TB - 1) / TB), TB, 0, stream>>>(x, W1, h, N, F, H);
    }

    {   // wave32 per edge (+ self loops)
        long long tot = ((long long)E + N) * 32;
        gcn_scatter64_kernel<<<(int)((tot + TB - 1) / TB), TB, 0, stream>>>(
            src, dst, deg, h, hagg, E, N);
    }

    gcn_gemm2_kernel<<<(N + TB - 1) / TB, TB, 0, stream>>>(hagg, b1, W2, g, N, H, C);

    {
        long long tot = (long long)E + N;
        gcn_scatterC_kernel<<<(int)((tot + TB - 1) / TB), TB, 0, stream>>>(
            src, dst, deg, g, out, E, N, C);
    }

    {
        long long tot = (long long)N * C;
        gcn_bias2_kernel<<<(int)((tot + TB - 1) / TB), TB, 0, stream>>>(out, b2, N, C);
    }
}